// ResidualMambaBlock_38886633898419
// MI455X (gfx1250) — compile-verified
//
#include <hip/hip_runtime.h>
#include <math.h>

// ---------------------------------------------------------------------------
// ResidualMambaBlock for MI455X (gfx1250, wave32, WMMA)
// fp32 end-to-end; GEMMs on v_wmma_f32_16x16x4_f32 (exact fp32 matrix pipe),
// double-buffered LDS with async global->LDS staging (ASYNCcnt path).
// ---------------------------------------------------------------------------

typedef __attribute__((ext_vector_type(2))) float v2f;
typedef __attribute__((ext_vector_type(4))) float v4f;
typedef __attribute__((ext_vector_type(8))) float v8f;
typedef __attribute__((ext_vector_type(4))) int   v4i;

#define D_MODEL 512
#define D_STATE 16
#define D_CONV  4
#define D_INNER 1024
#define DT_RANK 32
#define BATCH   8
#define SEQ     1024
#define NTOK    (BATCH * SEQ)   // 8192 rows for all GEMMs
#define LN_EPS  1e-5f

#define __AS1 __attribute__((address_space(1)))
#define __AS3 __attribute__((address_space(3)))

#if __has_builtin(__builtin_amdgcn_global_load_async_to_lds_b128)
#define USE_ASYNC_COPY 1
#else
#define USE_ASYNC_COPY 0
#endif

__device__ __forceinline__ void wait_async0() {
#if __has_builtin(__builtin_amdgcn_s_wait_asynccnt)
  __builtin_amdgcn_s_wait_asynccnt(0);
#else
  asm volatile("s_wait_asynccnt 0x0" ::: "memory");
#endif
}

// ---------------------------------------------------------------------------
// LayerNorm over last dim (512). One block (256 threads) per token row.
// ---------------------------------------------------------------------------
__global__ __launch_bounds__(256) void ln_kernel(
    const float* __restrict__ x, const float* __restrict__ g,
    const float* __restrict__ b, float* __restrict__ xn) {
  __shared__ float s_sum[8];
  __shared__ float s_sq[8];
  const int row = blockIdx.x;
  const int tid = threadIdx.x;
  const float* xr = x + (size_t)row * D_MODEL;

  float2 v = ((const float2*)xr)[tid];          // 256 threads * 2 = 512
  float sum = v.x + v.y;
  float sq  = v.x * v.x + v.y * v.y;
#pragma unroll
  for (int off = 16; off > 0; off >>= 1) {
    sum += __shfl_down(sum, off, 32);
    sq  += __shfl_down(sq,  off, 32);
  }
  const int lane = tid & 31, wv = tid >> 5;
  if (lane == 0) { s_sum[wv] = sum; s_sq[wv] = sq; }
  __syncthreads();
  if (wv == 0) {
    float s2 = (lane < 8) ? s_sum[lane] : 0.f;
    float q2 = (lane < 8) ? s_sq[lane]  : 0.f;
#pragma unroll
    for (int off = 4; off > 0; off >>= 1) {
      s2 += __shfl_down(s2, off, 32);
      q2 += __shfl_down(q2, off, 32);
    }
    if (lane == 0) { s_sum[0] = s2; s_sq[0] = q2; }
  }
  __syncthreads();
  const float mu   = s_sum[0] * (1.0f / D_MODEL);
  const float var  = s_sq[0] * (1.0f / D_MODEL) - mu * mu;
  const float rstd = rsqrtf(var + LN_EPS);

  float2 gg = ((const float2*)g)[tid];
  float2 bb = ((const float2*)b)[tid];
  float2 o;
  o.x = (v.x - mu) * rstd * gg.x + bb.x;
  o.y = (v.y - mu) * rstd * gg.y + bb.y;
  ((float2*)(xn + (size_t)row * D_MODEL))[tid] = o;
}

// ---------------------------------------------------------------------------
// Generic fp32 WMMA GEMM:  C[M,N] = A[M,K] * Bw[N,K]^T  (+ epilogue)
//   A row-major with leading dim lda; Bw row-major [N,K] (all weights are).
//   Block = 256 threads = 8 wave32; block tile 128(M) x 64(N); K chunk 32.
//   Double-buffered LDS, async global->LDS staging, 1 barrier per chunk.
// Epilogue: 0 = plain store, 1 = softplus(acc + bias[col]), 2 = acc + res.
// ---------------------------------------------------------------------------
#define LDS_STRIDE 36   // 36 dwords = 144 B: 16B-aligned rows, conflict-free

template <int EPI>
__global__ __launch_bounds__(256) void gemm_nt_wmma(
    const float* __restrict__ A, int lda,
    const float* __restrict__ Bw,
    float* __restrict__ C, int ldc, int K,
    const float* __restrict__ bias,
    const float* __restrict__ res, int ldr) {
  __shared__ float sA[2][128 * LDS_STRIDE];
  __shared__ float sB[2][64 * LDS_STRIDE];

  const int tid   = threadIdx.x;
  const int lane  = tid & 31;
  const int wv    = tid >> 5;            // 0..7
  const int blockM = blockIdx.y * 128;
  const int blockN = blockIdx.x * 64;

  // cooperative-load coords: 8 threads per row, float4 each -> 32 rows/pass
  const int col4 = (tid & 7) * 4;        // 0,4,...,28
  const int rowg = tid >> 3;             // 0..31

  v8f acc0 = {}, acc1 = {}, acc2 = {}, acc3 = {};

  const int mrow  = wv * 16 + (lane & 15);
  const int khalf = (lane >> 4) * 2;     // lanes 0-15: K pair {0,1}; 16-31: {2,3}
  const int nlo   = lane & 15;

#if !USE_ASYNC_COPY
  v4f ra[4];
  v4f rb[2];
#endif

  // ---- stage chunk kc into LDS buffer `buf` ----
  auto stage = [&](int kc, int buf) {
#if USE_ASYNC_COPY
#pragma unroll
    for (int r = 0; r < 4; ++r)
      __builtin_amdgcn_global_load_async_to_lds_b128(
          (__AS1 v4i*)(A + (size_t)(blockM + rowg + r * 32) * lda + kc + col4),
          (__AS3 v4i*)&sA[buf][(rowg + r * 32) * LDS_STRIDE + col4], 0, 0);
#pragma unroll
    for (int r = 0; r < 2; ++r)
      __builtin_amdgcn_global_load_async_to_lds_b128(
          (__AS1 v4i*)(Bw + (size_t)(blockN + rowg + r * 32) * K + kc + col4),
          (__AS3 v4i*)&sB[buf][(rowg + r * 32) * LDS_STRIDE + col4], 0, 0);
#else
    (void)buf;
#pragma unroll
    for (int r = 0; r < 4; ++r)
      ra[r] = *(const v4f*)(A + (size_t)(blockM + rowg + r * 32) * lda + kc + col4);
#pragma unroll
    for (int r = 0; r < 2; ++r)
      rb[r] = *(const v4f*)(Bw + (size_t)(blockN + rowg + r * 32) * K + kc + col4);
#endif
  };

#if !USE_ASYNC_COPY
  auto commit = [&](int buf) {
#pragma unroll
    for (int r = 0; r < 4; ++r)
      *(v4f*)(&sA[buf][(rowg + r * 32) * LDS_STRIDE + col4]) = ra[r];
#pragma unroll
    for (int r = 0; r < 2; ++r)
      *(v4f*)(&sB[buf][(rowg + r * 32) * LDS_STRIDE + col4]) = rb[r];
  };
#endif

  // ---- 8 WMMA steps of K=4 each on LDS buffer `buf` ----
  auto compute = [&](int buf) {
#pragma unroll
    for (int kk = 0; kk < 32; kk += 4) {
      v2f a  = *(const v2f*)(&sA[buf][mrow * LDS_STRIDE + kk + khalf]);
      v2f b0 = *(const v2f*)(&sB[buf][(0 * 16 + nlo) * LDS_STRIDE + kk + khalf]);
      v2f b1 = *(const v2f*)(&sB[buf][(1 * 16 + nlo) * LDS_STRIDE + kk + khalf]);
      v2f b2 = *(const v2f*)(&sB[buf][(2 * 16 + nlo) * LDS_STRIDE + kk + khalf]);
      v2f b3 = *(const v2f*)(&sB[buf][(3 * 16 + nlo) * LDS_STRIDE + kk + khalf]);
      acc0 = __builtin_amdgcn_wmma_f32_16x16x4_f32(false, a, false, b0, (short)0, acc0, false, false);
      acc1 = __builtin_amdgcn_wmma_f32_16x16x4_f32(false, a, false, b1, (short)0, acc1, false, false);
      acc2 = __builtin_amdgcn_wmma_f32_16x16x4_f32(false, a, false, b2, (short)0, acc2, false, false);
      acc3 = __builtin_amdgcn_wmma_f32_16x16x4_f32(false, a, false, b3, (short)0, acc3, false, false);
    }
  };

  // ---- software pipeline: fill buf0, then overlap fill(i+1) with math(i) ----
#if USE_ASYNC_COPY
  stage(0, 0);
  wait_async0();
#else
  stage(0, 0);
  commit(0);
#endif
  __syncthreads();

  int cur = 0;
  for (int kc = 32; kc < K; kc += 32) {
#if USE_ASYNC_COPY
    stage(kc, cur ^ 1);       // async fills run under the WMMA stream
    compute(cur);
    wait_async0();
#else
    stage(kc, cur ^ 1);       // global loads issue early...
    compute(cur);             // ...latency hidden under WMMA
    commit(cur ^ 1);
#endif
    __syncthreads();
    cur ^= 1;
  }
  compute(cur);

  // ---- epilogue: C/D layout = VGPR v holds M=v (lanes 0-15) / M=v+8 (16-31)
  const int mhi = (lane >> 4) * 8;
#pragma unroll
  for (int v = 0; v < 8; ++v) {
    const int grow = blockM + wv * 16 + mhi + v;
    float r0 = acc0[v], r1 = acc1[v], r2 = acc2[v], r3 = acc3[v];
    const int c0 = blockN + 0 * 16 + nlo;
    const int c1 = blockN + 1 * 16 + nlo;
    const int c2 = blockN + 2 * 16 + nlo;
    const int c3 = blockN + 3 * 16 + nlo;
    if (EPI == 1) {  // softplus(acc + bias[col])
      r0 += bias[c0]; r1 += bias[c1]; r2 += bias[c2]; r3 += bias[c3];
      r0 = (r0 > 20.f) ? r0 : log1pf(expf(r0));
      r1 = (r1 > 20.f) ? r1 : log1pf(expf(r1));
      r2 = (r2 > 20.f) ? r2 : log1pf(expf(r2));
      r3 = (r3 > 20.f) ? r3 : log1pf(expf(r3));
    } else if (EPI == 2) {  // residual add
      const float* rr = res + (size_t)grow * ldr;
      r0 += rr[c0]; r1 += rr[c1]; r2 += rr[c2]; r3 += rr[c3];
    }
    float* cr = C + (size_t)grow * ldc;
    cr[c0] = r0; cr[c1] = r1; cr[c2] = r2; cr[c3] = r3;
  }
}

// ---------------------------------------------------------------------------
// Causal depthwise conv1d (k=4) + bias + SiLU. One thread per (b,l,d).
// u = xz[:, 0:1024] with row stride 2048.
// ---------------------------------------------------------------------------
__global__ __launch_bounds__(256) void conv_silu_kernel(
    const float* __restrict__ xz, const float* __restrict__ w,
    const float* __restrict__ cb, float* __restrict__ uct) {
  const int idx = blockIdx.x * blockDim.x + threadIdx.x;   // b*2^20 + l*2^10 + d
  const int d = idx & (D_INNER - 1);
  const int l = (idx >> 10) & (SEQ - 1);
  const int b = idx >> 20;
  const float* up = xz + (size_t)b * SEQ * (2 * D_INNER);
  float acc = cb[d];
#pragma unroll
  for (int j = 0; j < D_CONV; ++j) {
    const int lj = l - (D_CONV - 1) + j;
    if (lj >= 0) acc += w[d * D_CONV + j] * up[(size_t)lj * (2 * D_INNER) + d];
  }
  const float sig = 1.f / (1.f + expf(-acc));
  uct[idx] = acc * sig;
}

// ---------------------------------------------------------------------------
// Selective scan. 32 blocks x 256 threads; lane owns one (b,d) channel,
// 16-state h in VGPRs. B/C are block-uniform per step (b, l uniform), so
// they lower to scalar-cache broadcast loads: no LDS, no barriers in the
// 1024-step serial loop. Fuses: y += u*D; gate y *= silu(z).
// ---------------------------------------------------------------------------
__global__ __launch_bounds__(256) void scan_kernel(
    const float* __restrict__ delta, const float* __restrict__ uct,
    const float* __restrict__ xdbl, const float* __restrict__ xz,
    const float* __restrict__ A_log, const float* __restrict__ Dp,
    float* __restrict__ yf) {
  const int b = blockIdx.x >> 2;                       // 4 blocks per batch
  const int d = ((blockIdx.x & 3) << 8) + threadIdx.x; // channel in [0,1024)

  float a[D_STATE];
#pragma unroll
  for (int s = 0; s < D_STATE; ++s) a[s] = -expf(A_log[d * D_STATE + s]);
  const float Dv = Dp[d];

  float h[D_STATE];
#pragma unroll
  for (int s = 0; s < D_STATE; ++s) h[s] = 0.f;

  const float* bc   = xdbl + (size_t)b * SEQ * (DT_RANK + 2 * D_STATE) + DT_RANK;
  const float* dptr = delta + (size_t)b * SEQ * D_INNER + d;
  const float* uptr = uct   + (size_t)b * SEQ * D_INNER + d;
  const float* zptr = xz    + (size_t)b * SEQ * (2 * D_INNER) + D_INNER + d;
  float*       yptr = yf    + (size_t)b * SEQ * D_INNER + d;

  for (int l = 0; l < SEQ; ++l) {
    // uniform addresses -> scalar (s_load) broadcast of the 32 B/C floats
    float Bv[D_STATE], Cv[D_STATE];
#pragma unroll
    for (int s = 0; s < D_STATE; ++s) {
      Bv[s] = bc[(size_t)l * (DT_RANK + 2 * D_STATE) + s];
      Cv[s] = bc[(size_t)l * (DT_RANK + 2 * D_STATE) + D_STATE + s];
    }
    const float dt = dptr[(size_t)l * D_INNER];
    const float ut = uptr[(size_t)l * D_INNER];
    const float du = dt * ut;
    float y = 0.f;
#pragma unroll
    for (int s = 0; s < D_STATE; ++s) {
      const float dA = expf(dt * a[s]);
      h[s] = dA * h[s] + du * Bv[s];
      y += h[s] * Cv[s];
    }
    y += ut * Dv;
    const float z = zptr[(size_t)l * (2 * D_INNER)];
    yptr[(size_t)l * D_INNER] = y * (z / (1.f + expf(-z)));
  }
}

// ---------------------------------------------------------------------------
// Host launcher
// ---------------------------------------------------------------------------
extern "C" void kernel_launch(void* const* d_in, const int* in_sizes, int n_in,
                              void* d_out, int out_size, void* d_ws, size_t ws_size,
                              hipStream_t stream) {
  (void)in_sizes; (void)n_in; (void)out_size; (void)ws_size;

  const float* x         = (const float*)d_in[0];   // [8,1024,512]
  const float* ln_g      = (const float*)d_in[1];   // [512]
  const float* ln_b      = (const float*)d_in[2];   // [512]
  const float* in_proj_w = (const float*)d_in[3];   // [2048,512]
  const float* conv_w    = (const float*)d_in[4];   // [1024,1,4]
  const float* conv_b    = (const float*)d_in[5];   // [1024]
  const float* x_proj_w  = (const float*)d_in[6];   // [64,1024]
  const float* dt_proj_w = (const float*)d_in[7];   // [1024,32]
  const float* dt_proj_b = (const float*)d_in[8];   // [1024]
  const float* A_log     = (const float*)d_in[9];   // [1024,16]
  const float* Dp        = (const float*)d_in[10];  // [1024]
  const float* out_proj_w= (const float*)d_in[11];  // [512,1024]
  float* out = (float*)d_out;                       // [8,1024,512]

  float* ws = (float*)d_ws;
  float* xn    = ws;                              // 8192*512
  float* xz    = xn    + (size_t)NTOK * D_MODEL;  // 8192*2048 (u | z)
  float* uct   = xz    + (size_t)NTOK * 2 * D_INNER;  // 8192*1024
  float* xdbl  = uct   + (size_t)NTOK * D_INNER;  // 8192*64 (dt | B | C)
  float* delta = xdbl  + (size_t)NTOK * (DT_RANK + 2 * D_STATE);  // 8192*1024
  float* yf    = delta + (size_t)NTOK * D_INNER;  // 8192*1024

  // 1) LayerNorm
  ln_kernel<<<NTOK, 256, 0, stream>>>(x, ln_g, ln_b, xn);

  // 2) in_proj: xz[8192,2048] = xn[8192,512] @ in_proj_w^T
  gemm_nt_wmma<0><<<dim3(2 * D_INNER / 64, NTOK / 128), 256, 0, stream>>>(
      xn, D_MODEL, in_proj_w, xz, 2 * D_INNER, D_MODEL, nullptr, nullptr, 0);

  // 3) depthwise causal conv + SiLU on u half
  conv_silu_kernel<<<(NTOK * D_INNER) / 256, 256, 0, stream>>>(xz, conv_w, conv_b, uct);

  // 4) x_proj: xdbl[8192,64] = uct @ x_proj_w^T
  gemm_nt_wmma<0><<<dim3(64 / 64, NTOK / 128), 256, 0, stream>>>(
      uct, D_INNER, x_proj_w, xdbl, DT_RANK + 2 * D_STATE, D_INNER,
      nullptr, nullptr, 0);

  // 5) delta = softplus(xdbl[:, :32] @ dt_proj_w^T + dt_proj_b)
  gemm_nt_wmma<1><<<dim3(D_INNER / 64, NTOK / 128), 256, 0, stream>>>(
      xdbl, DT_RANK + 2 * D_STATE, dt_proj_w, delta, D_INNER, DT_RANK,
      dt_proj_b, nullptr, 0);

  // 6) selective scan + D skip + SiLU(z) gating -> yf
  scan_kernel<<<(BATCH * D_INNER) / 256, 256, 0, stream>>>(
      delta, uct, xdbl, xz, A_log, Dp, yf);

  // 7) out_proj + residual: out = x + yf @ out_proj_w^T
  gemm_nt_wmma<2><<<dim3(D_MODEL / 64, NTOK / 128), 256, 0, stream>>>(
      yf, D_INNER, out_proj_w, out, D_MODEL, D_INNER, nullptr, x, D_MODEL);
}